// MultiHead_SelfAttention_88055419503384
// MI455X (gfx1250) — compile-verified
//
#include <hip/hip_runtime.h>
#include <math.h>

// Problem constants (B,N,D)=(8,512,512), H=8, DK=64
#define BATCH 8
#define SEQ   512
#define DIM   512
#define DK    64
#define QKV3  192            // 3*DK
#define SCALE 0.125f         // 1/sqrt(64)

typedef __attribute__((ext_vector_type(2))) float v2f;
typedef __attribute__((ext_vector_type(8))) float v8f;

// ---------------------------------------------------------------------------
// Stage 1: QKV = X(4096x512) @ Wqkv(512x192), written transposed as
// Qt/Kt/Vt[b][c][n] so the attention stage reads contiguous channel vectors.
// One wave per 16x16 output tile; K-loop of v_wmma_f32_16x16x4_f32.
// ---------------------------------------------------------------------------
__global__ __launch_bounds__(256) void qkv_gemm_kernel(
    const float* __restrict__ X, const float* __restrict__ Wqkv,
    float* __restrict__ Qt, float* __restrict__ Kt, float* __restrict__ Vt)
{
    const int tid  = threadIdx.x;
    const int lane = tid & 31;
    const int wave = tid >> 5;
    const int tile = blockIdx.x * 8 + wave;     // 0..3071
    const int JT   = QKV3 / 16;                 // 12 col tiles
    const int mt   = tile / JT;                 // 0..255  (row tile: 16 rows of b*512+n)
    const int jt   = tile % JT;
    const int row0 = mt * 16;
    const int j0   = jt * 16;
    const int lm   = lane & 15;                 // M (A) / N (B) index within tile
    const int lk   = lane >> 4;                 // K half select

    v8f acc = {0.f,0.f,0.f,0.f,0.f,0.f,0.f,0.f};

    const float* arow = X + (size_t)(row0 + lm) * DIM + 2 * lk;
    const float* bcol = Wqkv + j0 + lm;

    #pragma unroll 4
    for (int k0 = 0; k0 < DIM; k0 += 4) {
        // A frag: lane<16 -> {A[m][k0],A[m][k0+1]}, lane>=16 -> {A[m][k0+2],A[m][k0+3]}
        v2f a = *(const v2f*)(arow + k0);
        // B frag: rows k0+2*lk, k0+2*lk+1, column j0+lm
        const int kb = k0 + 2 * lk;
        v2f b;
        b.x = bcol[(size_t)kb * QKV3];
        b.y = bcol[(size_t)(kb + 1) * QKV3];
        acc = __builtin_amdgcn_wmma_f32_16x16x4_f32(
            false, a, false, b, (short)0, acc, false, false);
    }

    // D layout: VGPR r -> M = r + (lane>=16 ? 8 : 0), N = lm
    const int b_idx = row0 >> 9;                // row0 / 512
    const int n0    = (row0 & 511) + lk * 8;
    const int col   = j0 + lm;                  // 0..191 (tile never crosses 64-boundary)
    const int which = j0 >> 6;                  // 0=Q 1=K 2=V (uniform per tile)
    const int c     = col & 63;
    float* T = (which == 0) ? Qt : ((which == 1) ? Kt : Vt);
    float* base = T + ((size_t)b_idx * DK + c) * SEQ;
    #pragma unroll
    for (int r = 0; r < 8; ++r)
        base[n0 + r] = acc[r];
}

// ---------------------------------------------------------------------------
// Stage 2: per-channel rank-1 attention. One block per (b,c) pair.
// S[n,m] = q_n * k_m * scale  (rank-1) -> row max is analytic from kmax/kmin,
// softmax + V contraction fused, A never materialized (saves ~1 GB of HBM
// round-trips vs materializing the (B,DK,N,N) score tensor).
// Inner loop: ds_load_b128 K/V quads + 4 independent accumulator chains so
// the loop is bound by the transcendental pipe, not the FMA-chain latency.
// ---------------------------------------------------------------------------
__global__ __launch_bounds__(256) void attn_kernel(
    const float* __restrict__ Qt, const float* __restrict__ Kt,
    const float* __restrict__ Vt, float* __restrict__ SA)
{
    __shared__ __align__(16) float kv[SEQ];
    __shared__ __align__(16) float vv[SEQ];
    __shared__ float rmax[8];
    __shared__ float rmin[8];

    const int bc  = blockIdx.x;                 // b*64 + c
    const int tid = threadIdx.x;                // 0..255

    const float* kp = Kt + (size_t)bc * SEQ;
    const float* vp = Vt + (size_t)bc * SEQ;
    kv[tid]       = kp[tid];
    kv[tid + 256] = kp[tid + 256];
    vv[tid]       = vp[tid];
    vv[tid + 256] = vp[tid + 256];
    __syncthreads();

    // block-wide kmax/kmin (wave32 shuffle tree, then 8-wave combine)
    float kmx = fmaxf(kv[tid], kv[tid + 256]);
    float kmn = fminf(kv[tid], kv[tid + 256]);
    #pragma unroll
    for (int off = 16; off >= 1; off >>= 1) {
        kmx = fmaxf(kmx, __shfl_xor(kmx, off, 32));
        kmn = fminf(kmn, __shfl_xor(kmn, off, 32));
    }
    if ((tid & 31) == 0) { rmax[tid >> 5] = kmx; rmin[tid >> 5] = kmn; }
    __syncthreads();
    float KMAX = rmax[0], KMIN = rmin[0];
    #pragma unroll
    for (int i = 1; i < 8; ++i) {
        KMAX = fmaxf(KMAX, rmax[i]);
        KMIN = fminf(KMIN, rmin[i]);
    }

    const int b = bc >> 6;
    const int c = bc & 63;
    const float* qp = Qt + (size_t)bc * SEQ;
    const float4* kv4 = (const float4*)kv;
    const float4* vv4 = (const float4*)vv;

    #pragma unroll
    for (int r = 0; r < 2; ++r) {
        const int n = tid + r * 256;
        const float a    = qp[n] * SCALE;
        const float cmax = a * ((a >= 0.f) ? KMAX : KMIN);   // = max_m a*k_m
        float s0 = 0.f, s1 = 0.f, s2 = 0.f, s3 = 0.f;        // softmax denominator
        float p0 = 0.f, p1 = 0.f, p2 = 0.f, p3 = 0.f;        // numerator (dot with V)
        #pragma unroll 2
        for (int m4 = 0; m4 < SEQ / 4; ++m4) {
            const float4 k4 = kv4[m4];
            const float4 v4 = vv4[m4];
            const float e0 = __expf(fmaf(a, k4.x, -cmax));
            const float e1 = __expf(fmaf(a, k4.y, -cmax));
            const float e2 = __expf(fmaf(a, k4.z, -cmax));
            const float e3 = __expf(fmaf(a, k4.w, -cmax));
            s0 += e0; s1 += e1; s2 += e2; s3 += e3;
            p0 = fmaf(e0, v4.x, p0);
            p1 = fmaf(e1, v4.y, p1);
            p2 = fmaf(e2, v4.z, p2);
            p3 = fmaf(e3, v4.w, p3);
        }
        const float sum = (s0 + s1) + (s2 + s3);
        const float num = (p0 + p1) + (p2 + p3);
        SA[((size_t)b * SEQ + n) * DK + c] = num / sum;
    }
}

// ---------------------------------------------------------------------------
// Stage 3: Out(4096x512) = SA(4096x64) @ Wo(64x512). One wave per 16x16 tile,
// 16 v_wmma_f32_16x16x4_f32 per tile.
// ---------------------------------------------------------------------------
__global__ __launch_bounds__(256) void out_gemm_kernel(
    const float* __restrict__ SA, const float* __restrict__ Wo,
    float* __restrict__ Out)
{
    const int tid  = threadIdx.x;
    const int lane = tid & 31;
    const int wave = tid >> 5;
    const int tile = blockIdx.x * 8 + wave;     // 0..8191
    const int JT   = DIM / 16;                  // 32
    const int mt   = tile / JT;
    const int jt   = tile % JT;
    const int row0 = mt * 16;
    const int j0   = jt * 16;
    const int lm   = lane & 15;
    const int lk   = lane >> 4;

    v8f acc = {0.f,0.f,0.f,0.f,0.f,0.f,0.f,0.f};

    const float* arow = SA + (size_t)(row0 + lm) * DK + 2 * lk;
    const float* bcol = Wo + j0 + lm;

    #pragma unroll
    for (int k0 = 0; k0 < DK; k0 += 4) {
        v2f a = *(const v2f*)(arow + k0);
        const int kb = k0 + 2 * lk;
        v2f b;
        b.x = bcol[(size_t)kb * DIM];
        b.y = bcol[(size_t)(kb + 1) * DIM];
        acc = __builtin_amdgcn_wmma_f32_16x16x4_f32(
            false, a, false, b, (short)0, acc, false, false);
    }

    float* obase = Out + (size_t)(row0 + lk * 8) * DIM + j0 + lm;
    #pragma unroll
    for (int r = 0; r < 8; ++r)
        obase[(size_t)r * DIM] = acc[r];
}

// ---------------------------------------------------------------------------
extern "C" void kernel_launch(void* const* d_in, const int* in_sizes, int n_in,
                              void* d_out, int out_size, void* d_ws, size_t ws_size,
                              hipStream_t stream)
{
    const float* X    = (const float*)d_in[0];   // (8,512,512)
    const float* Wqkv = (const float*)d_in[1];   // (512,192)
    const float* Wo   = (const float*)d_in[2];   // (64,512)
    float* out = (float*)d_out;                  // (8,512,512)

    const size_t CHN = (size_t)BATCH * DK * SEQ; // 262144 floats per tensor
    float* Qt = (float*)d_ws;
    float* Kt = Qt + CHN;
    float* Vt = Kt + CHN;
    float* SA = Vt + CHN;                        // (B,N,DK) row-major

    // Stage 1: 256 row-tiles * 12 col-tiles = 3072 waves, 8 waves/block
    qkv_gemm_kernel<<<384, 256, 0, stream>>>(X, Wqkv, Qt, Kt, Vt);
    // Stage 2: one block per (b, c)
    attn_kernel<<<BATCH * DK, 256, 0, stream>>>(Qt, Kt, Vt, SA);
    // Stage 3: 256 row-tiles * 32 col-tiles = 8192 waves, 8 waves/block
    out_gemm_kernel<<<1024, 256, 0, stream>>>(SA, Wo, out);
}